// GCN_88502096101881
// MI455X (gfx1250) — compile-verified
//
#include <hip/hip_runtime.h>
#include <hip/hip_bf16.h>

#define N_NODES   100000
#define N_EDGES   640000
#define FEAT      128
#define HID       128
#define N_CLASSES 40
#define N_GRAPHS  128
#define BN_EPS    1e-5f

typedef float v2f __attribute__((ext_vector_type(2)));
typedef float v8f __attribute__((ext_vector_type(8)));

__device__ __forceinline__ void atomAddF(float* p, float v) {
    // lowers to global_atomic_add_f32 on AMD
    unsafeAtomicAdd(p, v);
}

// ---------------------------------------------------------------------------
// GEMM: D[r0:r0+16, 0:128] = A[r0:r0+16, 0:128] @ W[128,128] (+bias, relu)
// One block = 256 threads = 8 waves; wave w owns the 16x16 tile at cols w*16.
// A tile staged in LDS before a barrier -> safe to run IN PLACE (A == D).
// K=128 consumed as 32 x v_wmma_f32_16x16x4_f32.
// ---------------------------------------------------------------------------
__global__ void __launch_bounds__(256)
gemm128_wmma(const float* __restrict__ A, const float* __restrict__ W,
             float* __restrict__ D, int n_rows,
             const float* __restrict__ bias, int relu)
{
    __shared__ float As[16][132];  // +4 pad to dodge bank conflicts

    const int r0 = blockIdx.x * 16;

    // cooperative load of the 16x128 A tile (float4 per thread, 2 iters)
    for (int i = threadIdx.x; i < 16 * 32; i += 256) {
        const int rr  = i >> 5;
        const int cc4 = i & 31;
        int r = r0 + rr;
        float4 v = (r < n_rows) ? ((const float4*)(A + (size_t)r * 128))[cc4]
                                : make_float4(0.f, 0.f, 0.f, 0.f);
        As[rr][cc4 * 4 + 0] = v.x;
        As[rr][cc4 * 4 + 1] = v.y;
        As[rr][cc4 * 4 + 2] = v.z;
        As[rr][cc4 * 4 + 3] = v.w;
    }
    __syncthreads();

    const int wave = threadIdx.x >> 5;
    const int lane = threadIdx.x & 31;
    const int half = lane >> 4;   // 0: K pair {0,1}; 1: K pair {2,3}
    const int lrow = lane & 15;
    const int c0   = wave * 16;

    v8f acc = {};
#pragma unroll
    for (int k = 0; k < 32; ++k) {
        const int koff = 4 * k + 2 * half;
        v2f a, b;
        // A fragment (16x4 f32): lane row = lrow, two consecutive K values
        a.x = As[lrow][koff];
        a.y = As[lrow][koff + 1];
        // B fragment (4x16 f32): row K=koff(+1), col = c0 + lrow
        b.x = W[(size_t)koff * 128 + c0 + lrow];
        b.y = W[(size_t)(koff + 1) * 128 + c0 + lrow];
        acc = __builtin_amdgcn_wmma_f32_16x16x4_f32(
            /*neg_a=*/false, a, /*neg_b=*/false, b,
            /*c_mod=*/(short)0, acc, /*reuse_a=*/false, /*reuse_b=*/false);
    }

    const int col = c0 + lrow;
    const float bv = bias ? bias[col] : 0.0f;
#pragma unroll
    for (int m = 0; m < 8; ++m) {
        const int row = r0 + m + 8 * half;   // C/D layout: lanes16-31 hold M=8..15
        if (row < n_rows) {
            float v = acc[m] + bv;
            if (relu) v = fmaxf(v, 0.0f);
            D[(size_t)row * 128 + col] = v;
        }
    }
}

// ---------------------------------------------------------------------------
// degree / inverse-sqrt-degree
// ---------------------------------------------------------------------------
__global__ void deg_kernel(const int* __restrict__ rows, float* __restrict__ deg,
                           int n_edges)
{
    int e = blockIdx.x * blockDim.x + threadIdx.x;
    if (e < n_edges) atomAddF(&deg[rows[e]], 1.0f);
}

__global__ void dinv_kernel(float* __restrict__ deg, int n)
{
    int i = blockIdx.x * blockDim.x + threadIdx.x;
    if (i < n) deg[i] = rsqrtf(deg[i] + 1.0f);
}

// ---------------------------------------------------------------------------
// edge propagation: out[col] += dinv[row]*dinv[col] * h[row]   (wave per edge)
// ---------------------------------------------------------------------------
__global__ void __launch_bounds__(256)
prop_kernel(const float* __restrict__ h, float* __restrict__ out,
            const int* __restrict__ rows, const int* __restrict__ cols,
            const float* __restrict__ dinv, int n_edges)
{
    int e = blockIdx.x * 8 + (threadIdx.x >> 5);
    if (e >= n_edges) return;
    const int lane = threadIdx.x & 31;
    const int r = rows[e], c = cols[e];
    const float nrm = dinv[r] * dinv[c];
    float4 v = ((const float4*)(h + (size_t)r * 128))[lane];
    float* o = out + (size_t)c * 128 + lane * 4;
    atomAddF(o + 0, nrm * v.x);
    atomAddF(o + 1, nrm * v.y);
    atomAddF(o + 2, nrm * v.z);
    atomAddF(o + 3, nrm * v.w);
}

// ---------------------------------------------------------------------------
// batchnorm statistics + apply
// ---------------------------------------------------------------------------
__global__ void __launch_bounds__(256)
bn_stats(const float* __restrict__ x, float* __restrict__ sum,
         float* __restrict__ sumsq, int n)
{
    const int col = threadIdx.x & 127;
    float s = 0.f, s2 = 0.f;
    for (int r = blockIdx.x * 2 + (threadIdx.x >> 7); r < n; r += gridDim.x * 2) {
        float v = x[(size_t)r * 128 + col];
        s += v;
        s2 += v * v;
    }
    atomAddF(&sum[col], s);
    atomAddF(&sumsq[col], s2);
}

__global__ void bn_finalize(const float* __restrict__ sum,
                            const float* __restrict__ sumsq,
                            const float* __restrict__ g, const float* __restrict__ b,
                            float* __restrict__ scale, float* __restrict__ shift,
                            float inv_n)
{
    int c = threadIdx.x;
    float mu  = sum[c] * inv_n;
    float var = fmaxf(sumsq[c] * inv_n - mu * mu, 0.0f);
    float sc  = g[c] * rsqrtf(var + BN_EPS);
    scale[c] = sc;
    shift[c] = b[c] - mu * sc;
}

__global__ void __launch_bounds__(256)
bn_apply_relu(float* __restrict__ x, const float* __restrict__ scale,
              const float* __restrict__ shift, int total)
{
    int i = blockIdx.x * 256 + threadIdx.x;
    if (i < total) {
        int c = i & 127;
        x[i] = fmaxf(x[i] * scale[c] + shift[c], 0.0f);
    }
}

// ---------------------------------------------------------------------------
// global mean pool over graph segments
// ---------------------------------------------------------------------------
__global__ void __launch_bounds__(256)
pool_kernel(const float* __restrict__ h, const int* __restrict__ batch,
            float* __restrict__ sums, float* __restrict__ cnts, int n)
{
    int node = blockIdx.x * 8 + (threadIdx.x >> 5);
    if (node >= n) return;
    const int lane = threadIdx.x & 31;
    const int g = batch[node];
    float4 v = ((const float4*)(h + (size_t)node * 128))[lane];
    float* o = sums + (size_t)g * 128 + lane * 4;
    atomAddF(o + 0, v.x);
    atomAddF(o + 1, v.y);
    atomAddF(o + 2, v.z);
    atomAddF(o + 3, v.w);
    if (lane == 0) atomAddF(&cnts[g], 1.0f);
}

__global__ void pool_div(const float* __restrict__ sums,
                         const float* __restrict__ cnts, float* __restrict__ xg)
{
    int i = blockIdx.x * blockDim.x + threadIdx.x;
    if (i < N_GRAPHS * HID) xg[i] = sums[i] / fmaxf(cnts[i >> 7], 1.0f);
}

// ---------------------------------------------------------------------------
// classifier head (tiny): logits = y1 @ Wg2 + bg2 ; then log_softmax
// ---------------------------------------------------------------------------
__global__ void head2_kernel(const float* __restrict__ y1,
                             const float* __restrict__ Wg2,
                             const float* __restrict__ bg2,
                             float* __restrict__ logits)
{
    int i = blockIdx.x * blockDim.x + threadIdx.x;
    if (i >= N_GRAPHS * N_CLASSES) return;
    int g = i / N_CLASSES, c = i - g * N_CLASSES;
    float s = bg2[c];
    for (int k = 0; k < HID; ++k) s += y1[g * HID + k] * Wg2[k * N_CLASSES + c];
    logits[i] = s;
}

__global__ void logsoftmax_kernel(const float* __restrict__ logits,
                                  float* __restrict__ out)
{
    int g = threadIdx.x;  // 128 rows
    const float* r = logits + g * N_CLASSES;
    float m = -3.402823466e38f;
    for (int c = 0; c < N_CLASSES; ++c) m = fmaxf(m, r[c]);
    float s = 0.f;
    for (int c = 0; c < N_CLASSES; ++c) s += __expf(r[c] - m);
    float ls = __logf(s);
    for (int c = 0; c < N_CLASSES; ++c) out[g * N_CLASSES + c] = r[c] - m - ls;
}

// ---------------------------------------------------------------------------
extern "C" void kernel_launch(void* const* d_in, const int* in_sizes, int n_in,
                              void* d_out, int out_size, void* d_ws, size_t ws_size,
                              hipStream_t stream)
{
    (void)in_sizes; (void)n_in; (void)out_size; (void)ws_size;

    const float* x    = (const float*)d_in[0];
    const int*   ei   = (const int*)d_in[1];
    const int*   row  = ei;
    const int*   col  = ei + N_EDGES;
    const int*   batch= (const int*)d_in[2];
    const float* W0   = (const float*)d_in[3];
    const float* g0   = (const float*)d_in[4];
    const float* b0   = (const float*)d_in[5];
    const float* W1   = (const float*)d_in[6];
    const float* g1   = (const float*)d_in[7];
    const float* b1   = (const float*)d_in[8];
    const float* W2   = (const float*)d_in[9];
    const float* Wg1  = (const float*)d_in[10];
    const float* bg1  = (const float*)d_in[11];
    const float* Wg2  = (const float*)d_in[12];
    const float* bg2  = (const float*)d_in[13];

    const size_t NB = (size_t)N_NODES * HID * sizeof(float);   // 51.2 MB

    char* ws = (char*)d_ws;
    size_t off = 0;
    auto carve = [&](size_t bytes) -> float* {
        float* p = (float*)(ws + off);
        off = (off + bytes + 255) & ~(size_t)255;
        return p;
    };
    float* buf1    = carve(NB);
    float* buf2    = carve(NB);
    float* dinv    = carve((size_t)N_NODES * sizeof(float));
    float* bnsum   = carve(128 * sizeof(float));   // contiguous with bnsq
    float* bnsq    = carve(128 * sizeof(float));
    float* scale   = carve(128 * sizeof(float));
    float* shift   = carve(128 * sizeof(float));
    float* poolsum = carve((size_t)N_GRAPHS * HID * sizeof(float)); // contiguous w/ cnts
    float* cnts    = carve(N_GRAPHS * sizeof(float));
    float* xg      = carve((size_t)N_GRAPHS * HID * sizeof(float));
    float* y1      = carve((size_t)N_GRAPHS * HID * sizeof(float));
    float* logits  = carve((size_t)N_GRAPHS * N_CLASSES * sizeof(float));

    const int gemmBlocks = N_NODES / 16;       // 6250 (N divisible by 16)
    const int propBlocks = N_EDGES / 8;        // 80000
    const int elemBlocks = (N_NODES * HID) / 256; // 50000

    // degrees -> dinv (in place)
    hipMemsetAsync(dinv, 0, (size_t)N_NODES * sizeof(float), stream);
    deg_kernel<<<(N_EDGES + 255) / 256, 256, 0, stream>>>(row, dinv, N_EDGES);
    dinv_kernel<<<(N_NODES + 255) / 256, 256, 0, stream>>>(dinv, N_NODES);

    // ---- layer 0 ----
    gemm128_wmma<<<gemmBlocks, 256, 0, stream>>>(x, W0, buf1, N_NODES, nullptr, 0);
    hipMemsetAsync(buf2, 0, NB, stream);
    prop_kernel<<<propBlocks, 256, 0, stream>>>(buf1, buf2, row, col, dinv, N_EDGES);
    hipMemsetAsync(bnsum, 0, 1024, stream);  // bnsum + bnsq
    bn_stats<<<1000, 256, 0, stream>>>(buf2, bnsum, bnsq, N_NODES);
    bn_finalize<<<1, 128, 0, stream>>>(bnsum, bnsq, g0, b0, scale, shift, 1.0f / N_NODES);
    bn_apply_relu<<<elemBlocks, 256, 0, stream>>>(buf2, scale, shift, N_NODES * HID);

    // ---- layer 1 ----
    gemm128_wmma<<<gemmBlocks, 256, 0, stream>>>(buf2, W1, buf2, N_NODES, nullptr, 0); // in place
    hipMemsetAsync(buf1, 0, NB, stream);
    prop_kernel<<<propBlocks, 256, 0, stream>>>(buf2, buf1, row, col, dinv, N_EDGES);
    hipMemsetAsync(bnsum, 0, 1024, stream);
    bn_stats<<<1000, 256, 0, stream>>>(buf1, bnsum, bnsq, N_NODES);
    bn_finalize<<<1, 128, 0, stream>>>(bnsum, bnsq, g1, b1, scale, shift, 1.0f / N_NODES);
    bn_apply_relu<<<elemBlocks, 256, 0, stream>>>(buf1, scale, shift, N_NODES * HID);

    // ---- layer 2 ----
    gemm128_wmma<<<gemmBlocks, 256, 0, stream>>>(buf1, W2, buf1, N_NODES, nullptr, 0); // in place
    hipMemsetAsync(buf2, 0, NB, stream);
    prop_kernel<<<propBlocks, 256, 0, stream>>>(buf1, buf2, row, col, dinv, N_EDGES);

    // ---- global mean pool ----
    hipMemsetAsync(poolsum, 0, (size_t)N_GRAPHS * HID * sizeof(float) + 256, stream); // + cnts
    pool_kernel<<<N_NODES / 8, 256, 0, stream>>>(buf2, batch, poolsum, cnts, N_NODES);
    pool_div<<<(N_GRAPHS * HID + 255) / 256, 256, 0, stream>>>(poolsum, cnts, xg);

    // ---- head ----
    gemm128_wmma<<<N_GRAPHS / 16, 256, 0, stream>>>(xg, Wg1, y1, N_GRAPHS, bg1, 1);
    head2_kernel<<<(N_GRAPHS * N_CLASSES + 255) / 256, 256, 0, stream>>>(y1, Wg2, bg2, logits);
    logsoftmax_kernel<<<1, 128, 0, stream>>>(logits, (float*)d_out);
}